// TSNE_48352741818804
// MI455X (gfx1250) — compile-verified
//
#include <hip/hip_runtime.h>
#include <math.h>

typedef __attribute__((ext_vector_type(2))) float v2f;
typedef __attribute__((ext_vector_type(8))) float v8f;

#define N_POINTS 8192
#define TILE 16
#define N_TILES (N_POINTS / TILE)            // 512
#define JMAX 257                              // j = 0..256 (symmetry rotation)
#define K1_BLOCKS 1024
#define K1_THREADS 256                        // 8 waves/block -> 8192 waves total

// Pre-packed WMMA operands (written by tsne_prep every launch; 256 KB, L2-resident).
//   g_packA[p] = ( x,  y,  |p|^2, 1 )      (A rows:  lanes<16 read .xy, lanes>=16 read .zw)
//   g_packB[p] = (-2x, -2y, 1,   |p|^2 )   (B cols:  same split)
__device__ float g_packA[N_POINTS * 4];
__device__ float g_packB[N_POINTS * 4];

__global__ void tsne_init(float* acc) {
    if (threadIdx.x < 3) acc[threadIdx.x] = 0.0f;
}

__global__ __launch_bounds__(256)
void tsne_prep(const float* __restrict__ logits) {
    const int p = blockIdx.x * 256 + threadIdx.x;
    if (p < N_POINTS) {
        const float x  = logits[2 * p];
        const float y  = logits[2 * p + 1];
        const float sq = x * x + y * y;
        reinterpret_cast<float4*>(g_packA)[p] = make_float4(x, y, sq, 1.0f);
        reinterpret_cast<float4*>(g_packB)[p] = make_float4(-2.0f * x, -2.0f * y, 1.0f, sq);
    }
}

// Partition function via symmetry: tile (rt, ct=(rt+j)&511), j in [0,256].
//   j==0   : diagonal tiles, weight 1
//   j==256 : each unordered pair enumerated twice, weight 1
//   else   : each unordered pair enumerated once, weight 2
// One V_WMMA_F32_16X16X4_F32 yields a 16x16 tile of squared distances:
//   D[m,n] = |p_m|^2 + |p_n|^2 - 2 x_m.x_n
__global__ __launch_bounds__(K1_THREADS)
void tsne_partition(float* __restrict__ acc) {
    const int lane = threadIdx.x & 31;
    // Force wave-uniform values onto the scalar path: loop control becomes
    // SALU + s_cbranch and EXEC stays all-ones (required for WMMA).
    const int wid = __builtin_amdgcn_readfirstlane(
        blockIdx.x * (K1_THREADS / 32) + (threadIdx.x >> 5));
    const int rt   = wid >> 4;               // row tile 0..511
    const int g    = wid & 15;               // j-stride group
    const int half = lane >> 4;              // K-half selector
    const int l    = lane & 15;              // M/N within tile

    const int loff = 4 * l + 2 * half;       // lane offset within a 64-float tile block
    const v2f a = *reinterpret_cast<const v2f*>(g_packA + 64 * rt + loff);

    float sum1 = 0.0f;                       // weight-1 tiles (j==0, j==256)
    float sum2 = 0.0f;                       // weight-2 tiles

    // Software-pipelined by one tile: prefetch next B before consuming current.
    int j = g;
    int ct = (rt + j) & (N_TILES - 1);
    v2f b = *reinterpret_cast<const v2f*>(g_packB + 64 * ct + loff);

    while (true) {
        const int jn = j + 16;
        v2f bn;
        if (jn < JMAX) {
            const int ctn = (rt + jn) & (N_TILES - 1);
            bn = *reinterpret_cast<const v2f*>(g_packB + 64 * ctn + loff);
        }

        v8f c = {};
        c = __builtin_amdgcn_wmma_f32_16x16x4_f32(
            /*neg_a=*/false, a, /*neg_b=*/false, b,
            /*c_mod=*/(short)0, c, /*reuse_a=*/false, /*reuse_b=*/false);

        // Two independent accumulation chains to overlap trans-pipe latency.
        float t0 = 0.0f, t1 = 0.0f;
#pragma unroll
        for (int r = 0; r < 4; ++r) {
            t0 += __builtin_amdgcn_rcpf(1.0f + c[2 * r]);
            t1 += __builtin_amdgcn_rcpf(1.0f + c[2 * r + 1]);
        }
        const float t = t0 + t1;

        if (j == 0 || j == 256) sum1 += t;   // scalar branch (j is uniform)
        else                    sum2 += t;

        if (jn >= JMAX) break;
        j = jn;
        b = bn;
    }

    float sum = 2.0f * sum2 + sum1;
    for (int off = 16; off > 0; off >>= 1)
        sum += __shfl_xor(sum, off, 32);
    if (lane == 0) atomicAdd(&acc[0], sum);
}

// Pair KL partial sums:
//   acc[1] += sum pij * (log(pij) + log(2 + ||xi - xj||^2))
//   acc[2] += sum pij
__global__ __launch_bounds__(256)
void tsne_pairs(const float* __restrict__ pij, const int* __restrict__ ii,
                const int* __restrict__ jj, const float* __restrict__ logits,
                float* __restrict__ acc, int npairs) {
    const int e = blockIdx.x * 256 + threadIdx.x;
    float t = 0.0f, p = 0.0f;
    if (e < npairs) {
        p = pij[e];
        const int i = ii[e];
        const int j = jj[e];
        const float dx = logits[2 * i]     - logits[2 * j];
        const float dy = logits[2 * i + 1] - logits[2 * j + 1];
        const float s  = 2.0f + dx * dx + dy * dy;   // D + ||xi-xj||^2, D==2
        t = p * (logf(p) + logf(s));                 // -log(num) == log(s)
    }
    for (int off = 16; off > 0; off >>= 1) {
        t += __shfl_xor(t, off, 32);
        p += __shfl_xor(p, off, 32);
    }
    if ((threadIdx.x & 31) == 0) {
        atomicAdd(&acc[1], t);
        atomicAdd(&acc[2], p);
    }
}

// KL = S1 + log(part) * S2, part = (full N^2 sum) - N  (diagonal removal)
__global__ void tsne_finalize(const float* __restrict__ acc, float* __restrict__ out) {
    const float part = acc[0] - (float)N_POINTS;
    out[0] = acc[1] + logf(part) * acc[2];
}

extern "C" void kernel_launch(void* const* d_in, const int* in_sizes, int n_in,
                              void* d_out, int out_size, void* d_ws, size_t ws_size,
                              hipStream_t stream) {
    const float* pij    = (const float*)d_in[0];
    const int*   ii     = (const int*)d_in[1];
    const int*   jj     = (const int*)d_in[2];
    const float* logits = (const float*)d_in[3];
    float* acc = (float*)d_ws;      // [0]=partition sum, [1]=pair term, [2]=sum pij
    float* out = (float*)d_out;
    const int npairs = in_sizes[0];

    tsne_init<<<1, 32, 0, stream>>>(acc);
    tsne_prep<<<(N_POINTS + 255) / 256, 256, 0, stream>>>(logits);
    tsne_partition<<<K1_BLOCKS, K1_THREADS, 0, stream>>>(acc);
    tsne_pairs<<<(npairs + 255) / 256, 256, 0, stream>>>(pij, ii, jj, logits, acc, npairs);
    tsne_finalize<<<1, 1, 0, stream>>>(acc, out);
}